// MyModel_87522843560001
// MI455X (gfx1250) — compile-verified
//
#include <hip/hip_runtime.h>

typedef __attribute__((ext_vector_type(2))) float v2f;
typedef __attribute__((ext_vector_type(8))) float v8f;

#define TPB    256
#define BQ     4            // elements per thread
#define EPB    (TPB * BQ)   // 1024 elements per block
#define MAX_IT 10
#define REG0   1e-20f
#define MULT   10.0f

// ---------------------------------------------------------------------------
// Wave-wide sum of 4 values/lane (pre-folded to q01,q23) via chained
// V_WMMA_F32_16X16X4_F32.  WMMA#1: A[m][k] = {q01(m),q23(m),q01(m+16),q23(m+16)},
// B = ones  ->  D1[m][n] = s(m) (16 row sums, replicated over N).
// WMMA#2..5: A = (d1[2r],d1[2r+1]), B = ones, accumulating C ->
// each adds s(2r)+s(2r+1)+s(2r+8)+s(2r+9); the four cover s(0..15) exactly
// once -> result = full 32-lane x 4-value total, broadcast to every lane.
// Fixed hardware accumulation order => deterministic across replays.
// ---------------------------------------------------------------------------
__device__ __forceinline__ float wmma_wave_sum(float q01, float q23) {
  v2f a;    a.x = q01;  a.y = q23;
  v2f ones; ones.x = 1.0f; ones.y = 1.0f;
  v8f zero = {0.f, 0.f, 0.f, 0.f, 0.f, 0.f, 0.f, 0.f};

  v8f d1 = __builtin_amdgcn_wmma_f32_16x16x4_f32(false, a, false, ones,
                                                 (short)0, zero, false, false);
  v8f acc = zero;
  v2f p;
  p.x = d1[0]; p.y = d1[1];
  acc = __builtin_amdgcn_wmma_f32_16x16x4_f32(false, p, false, ones,
                                              (short)0, acc, false, false);
  p.x = d1[2]; p.y = d1[3];
  acc = __builtin_amdgcn_wmma_f32_16x16x4_f32(false, p, false, ones,
                                              (short)0, acc, false, false);
  p.x = d1[4]; p.y = d1[5];
  acc = __builtin_amdgcn_wmma_f32_16x16x4_f32(false, p, false, ones,
                                              (short)0, acc, false, false);
  p.x = d1[6]; p.y = d1[7];
  acc = __builtin_amdgcn_wmma_f32_16x16x4_f32(false, p, false, ones,
                                              (short)0, acc, false, false);
  return acc[0];   // total, present in every lane
}

// Closed-form solve of (JTJ + lam*I) u = JTv for the arrowhead system.
__device__ __forceinline__ void lm_delta(float x0, float x1,
                                         float y0, float y1, float y2,
                                         float lam,
                                         float& ux0, float& ux1,
                                         float& uy0, float& uy1, float& uy2) {
  float S1 = y0 + y1 + y2;
  float S2 = y0 * y0 + y1 * y1 + y2 * y2;
  float Q  = x0 * x0 + x1 * x1;
  float X  = x0 + x1;
  float a  = 3.0f + S2 + lam;       // diag of x-block
  float b  = 2.0f + Q + lam;        // diag of y-block
  float c0 = 1.0f + x0 * y0, c1 = 1.0f + x0 * y1, c2 = 1.0f + x0 * y2;
  float d0 = 1.0f + x1 * y0, d1 = 1.0f + x1 * y1, d2 = 1.0f + x1 * y2;
  float gx0 = x0 * (3.0f + S2) + S1;
  float gx1 = x1 * (3.0f + S2) + S1;
  float gy0 = X + y0 * (2.0f + Q);
  float gy1 = X + y1 * (2.0f + Q);
  float gy2 = X + y2 * (2.0f + Q);
  float invb = 1.0f / b;
  float cc = c0 * c0 + c1 * c1 + c2 * c2;
  float cd = c0 * d0 + c1 * d1 + c2 * d2;
  float dd = d0 * d0 + d1 * d1 + d2 * d2;
  float M00 = a - cc * invb;
  float M01 = -cd * invb;
  float M11 = a - dd * invb;
  float p0 = gx0 - (c0 * gy0 + c1 * gy1 + c2 * gy2) * invb;
  float p1 = gx1 - (d0 * gy0 + d1 * gy1 + d2 * gy2) * invb;
  float invdet = 1.0f / (M00 * M11 - M01 * M01);
  ux0 = (M11 * p0 - M01 * p1) * invdet;
  ux1 = (M00 * p1 - M01 * p0) * invdet;
  uy0 = (gy0 - c0 * ux0 - d0 * ux1) * invb;
  uy1 = (gy1 - c1 * ux0 - d1 * ux1) * invb;
  uy2 = (gy2 - c2 * ux0 - d2 * ux1) * invb;
}

__device__ __forceinline__ float resid_ssq(float x0, float x1,
                                           float y0, float y1, float y2) {
  float s = 0.0f, t;
  t = x0 + y0; s += t * t;  t = x1 + y0; s += t * t;
  t = x0 + y1; s += t * t;  t = x1 + y1; s += t * t;
  t = x0 + y2; s += t * t;  t = x1 + y2; s += t * t;
  t = x0 * y0; s += t * t;  t = x1 * y0; s += t * t;
  t = x0 * y1; s += t * t;  t = x1 * y1; s += t * t;
  t = x0 * y2; s += t * t;  t = x1 * y2; s += t * t;
  return s;
}

// One launch per LM iteration K=0..10.
//  P layout: [MAX_IT+1][nblk]; row 0 = initial obj partials, row j+1 =
//  candidate-obj partials of iteration j (written by kernel j).
//  Kernel K reads rows 0..K (all written by earlier kernels this call),
//  replays the accept/reg chain, replays K state updates from the raw
//  inputs, then (K<10) emits row K+1, or (K==10) writes the result.
__global__ __launch_bounds__(TPB)
void lm_iter_kernel(const float* __restrict__ xin,
                    const float* __restrict__ yin,
                    float* __restrict__ P,
                    float* __restrict__ out,
                    int K, int nblk, int B) {
  __shared__ float sS[MAX_IT + 1];
  __shared__ float wpart[2][TPB / 32];

  const int tid  = threadIdx.x;
  const int lane = tid & 31;
  const int wid  = tid >> 5;

  // --- cross-block sums S_j = sum_b P[j][b], identical & deterministic in
  //     every block (fixed strided order + fixed shuffle tree) ---
  if (K > 0 && wid == 0) {
    for (int j = 0; j <= K; ++j) {
      float v = 0.0f;
      for (int b = lane; b < nblk; b += 32) v += P[j * nblk + b];
#pragma unroll
      for (int off = 16; off >= 1; off >>= 1) v += __shfl_xor(v, off, 32);
      if (lane == 0) sS[j] = v;
    }
  }
  __syncthreads();

  // --- load the 4 elements this thread owns ---
  float X0[BQ], X1[BQ], Y0[BQ], Y1[BQ], Y2[BQ];
  const int base = blockIdx.x * EPB + tid;
#pragma unroll
  for (int r = 0; r < BQ; ++r) {
    int e = base + r * TPB;
    X0[r] = xin[2 * e + 0];
    X1[r] = xin[2 * e + 1];
    Y0[r] = yin[3 * e + 0];
    Y1[r] = yin[3 * e + 1];
    Y2[r] = yin[3 * e + 2];
  }

  // --- replay accept chain + state through iterations 0..K-1 ---
  float reg = REG0;
  float obj = (K > 0) ? sS[0] : 0.0f;
  for (int j = 0; j < K && j < MAX_IT; ++j) {
    bool accept = sS[j + 1] < obj;   // NaN candidate => rejected (matches ref)
#pragma unroll
    for (int r = 0; r < BQ; ++r) {
      float ux0, ux1, uy0, uy1, uy2;
      lm_delta(X0[r], X1[r], Y0[r], Y1[r], Y2[r], reg, ux0, ux1, uy0, uy1, uy2);
      if (accept) {
        X0[r] -= ux0; X1[r] -= ux1;
        Y0[r] -= uy0; Y1[r] -= uy1; Y2[r] -= uy2;
      }
    }
    if (accept) obj = sS[j + 1]; else reg *= MULT;
  }

  if (K >= MAX_IT) {   // uniform branch: write minimized variables
#pragma unroll
    for (int r = 0; r < BQ; ++r) {
      int e = base + r * TPB;
      out[2 * e + 0] = X0[r];
      out[2 * e + 1] = X1[r];
      out[2 * B + 3 * e + 0] = Y0[r];
      out[2 * B + 3 * e + 1] = Y1[r];
      out[2 * B + 3 * e + 2] = Y2[r];
    }
    return;
  }

  // --- candidate for iteration K + its squared-residual partial ---
  float qc[BQ];
#pragma unroll
  for (int r = 0; r < BQ; ++r) {
    float ux0, ux1, uy0, uy1, uy2;
    lm_delta(X0[r], X1[r], Y0[r], Y1[r], Y2[r], reg, ux0, ux1, uy0, uy1, uy2);
    qc[r] = resid_ssq(X0[r] - ux0, X1[r] - ux1,
                      Y0[r] - uy0, Y1[r] - uy1, Y2[r] - uy2);
  }
  float tot = wmma_wave_sum(qc[0] + qc[1], qc[2] + qc[3]);
  if (lane == 0) wpart[0][wid] = tot;

  if (K == 0) {        // also emit the initial objective partials
    float qi[BQ];
#pragma unroll
    for (int r = 0; r < BQ; ++r)
      qi[r] = resid_ssq(X0[r], X1[r], Y0[r], Y1[r], Y2[r]);
    float tot0 = wmma_wave_sum(qi[0] + qi[1], qi[2] + qi[3]);
    if (lane == 0) wpart[1][wid] = tot0;
  }
  __syncthreads();

  if (tid == 0) {
    float s = 0.0f;
#pragma unroll
    for (int w = 0; w < TPB / 32; ++w) s += wpart[0][w];
    P[(K + 1) * nblk + blockIdx.x] = s;
  }
  if (K == 0 && tid == 32) {
    float s = 0.0f;
#pragma unroll
    for (int w = 0; w < TPB / 32; ++w) s += wpart[1][w];
    P[blockIdx.x] = s;
  }
}

extern "C" void kernel_launch(void* const* d_in, const int* in_sizes, int n_in,
                              void* d_out, int out_size, void* d_ws, size_t ws_size,
                              hipStream_t stream) {
  const float* x = (const float*)d_in[0];   // (B,1,2) fp32
  const float* y = (const float*)d_in[1];   // (B,3,1) fp32
  float* out = (float*)d_out;               // 2B floats (x) then 3B floats (y)
  float* P   = (float*)d_ws;                // [MAX_IT+1][nblk] partials

  const int B    = in_sizes[0] / 2;         // 262144
  const int nblk = B / EPB;                 // 256 blocks (B divisible by 1024)

  for (int K = 0; K <= MAX_IT; ++K) {
    lm_iter_kernel<<<nblk, TPB, 0, stream>>>(x, y, P, out, K, nblk, B);
  }
}